// InternalCoordinateDecoder_33672543601317
// MI455X (gfx1250) — compile-verified
//
#include <hip/hip_runtime.h>
#include <hip/hip_bf16.h>
#include <cstdint>

// InternalCoordinateDecoder for MI455X (gfx1250, wave32).
//
// Math: the reference chain is planar (n stays (0,0,+/-1)); each step is a
// 2x2 orthogonal frame update depending only on alpha_l. We run an
// associative scan over (A, v) with combine
//   (A1,v1) o (A2,v2) = (A1*A2, v1 + A1*v2),
// A = [[p,-d*q],[q,d*p]] (d=+/-1) stored as (p,q,d), v=(x,y).
// Coordinates: q_l = (1.6 + 1.6*S_y, 1.6 - 1.6*S_x, 0), seeds for l<3.
//
// Mapping: one 256-thread block (8 wave32 waves) per batch row; TDM
// (tensor_load_to_lds) DMA-gathers the strided alpha channel (2048 x 4B,
// stride 8B) into contiguous LDS; hierarchical scan: thread-serial ->
// wave32 shuffle -> cross-wave LDS.

#if defined(__gfx1250__) && __has_builtin(__builtin_amdgcn_tensor_load_to_lds)
#define ICD_USE_TDM 1
#if __has_include(<hip/amd_detail/amd_gfx1250_TDM.h>)
#define ICD_TDM_6ARG 1
#endif
#endif

namespace {

constexpr int   kL    = 2048;
constexpr int   kTPB  = 256;
constexpr int   kEPT  = kL / kTPB;       // 8 elements per thread
constexpr float kBL   = 1.6f;
constexpr float kD2R  = 0.017453292519943295f;

typedef unsigned int v4u __attribute__((ext_vector_type(4)));
typedef int          v4i __attribute__((ext_vector_type(4)));
typedef int          v8i __attribute__((ext_vector_type(8)));

struct S {
  float p, q, d, x, y;  // matrix [[p,-d q],[q,d p]], vector (x,y)
};

__device__ __forceinline__ S ident() {
  S r; r.p = 1.f; r.q = 0.f; r.d = 1.f; r.x = 0.f; r.y = 0.f; return r;
}

// combine: a (earlier elements) followed by b (later elements)
__device__ __forceinline__ S comb(const S a, const S b) {
  const float dq = a.d * a.q;
  const float dp = a.d * a.p;
  S r;
  r.p = a.p * b.p - dq * b.q;
  r.q = a.q * b.p + dp * b.q;
  r.d = a.d * b.d;
  r.x = a.x + a.p * b.x - dq * b.y;
  r.y = a.y + a.q * b.x + dp * b.y;
  return r;
}

__device__ __forceinline__ S elem(float alpha_deg, int l) {
  if (l < 3) return ident();
  const float a = alpha_deg * kD2R;
  float si, c;
  __sincosf(a, &si, &c);
  const float s = (si >= 0.f) ? 1.f : -1.f;
  S r; r.p = -c; r.q = -si; r.d = -s; r.x = c; r.y = si;
  return r;
}

__device__ __forceinline__ S shfl_up_state(const S v, int off) {
  S r;
  r.p = __shfl_up(v.p, off, 32);
  r.q = __shfl_up(v.q, off, 32);
  r.d = __shfl_up(v.d, off, 32);
  r.x = __shfl_up(v.x, off, 32);
  r.y = __shfl_up(v.y, off, 32);
  return r;
}

}  // namespace

__global__ __launch_bounds__(kTPB) void icd_scan_kernel(
    const float* __restrict__ torsions,       // [B, L, 2] fp32
    const unsigned char* __restrict__ mask,   // [B, L] bool (1 byte)
    float* __restrict__ out) {                // [B, L, 3] fp32
  const int b    = blockIdx.x;
  const int tid  = threadIdx.x;
  const int lane = tid & 31;
  const int wid  = tid >> 5;                  // 0..7
  const int l0   = tid * kEPT;

  const float* row = torsions + (size_t)b * kL * 2;   // channel-0 base
  const unsigned char* mrow = mask + (size_t)b * kL;

  // Probe-confirmed gfx1250 prefetch path (global_prefetch_b8).
  __builtin_prefetch(mrow + l0, 0, 0);

  float alpha[kEPT];

#if defined(ICD_USE_TDM)
  // ---- TDM: DMA-gather the strided alpha channel into contiguous LDS ----
  // 2-D descriptor: data_size=4B (code 2), lines of tile_dim0=1 element,
  // tile_dim1=2048 lines, tensor_dim0_stride=2 elements (8 bytes) -> packs
  // torsions[b, :, 0] into abuf[0..2047].
  __shared__ float abuf[kL];                  // 8 KB
  if (wid == 0) {
    const uint64_t ga  = (uint64_t)(uintptr_t)row;
    const uint32_t lds = (uint32_t)(uintptr_t)&abuf[0];
    // D# group 0: count=1 | lds_addr | global_addr[56:0] | type=2
    const v4u g0 = { 1u, lds, (uint32_t)ga,
                     (uint32_t)((ga >> 32) & 0x01FFFFFFu) | (2u << 30) };
    // D# group 1: workgroup_mask=0, data_size=2 (4B), no barrier/iterate/pad,
    // tensor_dim0=1, tensor_dim1=2048, tile_dim0=1, tile_dim1=2048,
    // tile_dim2=0, tensor_dim0_stride=2, tensor_dim1_stride=0.
    const v8i g1 = { (int)0x00020000,   // data_size=2<<16
                     (int)0x00010000,   // tensor_dim0=1 (bits 63:48)
                     (int)0x08000000,   // tensor_dim1=2048 (bits 95:80)
                     (int)0x00010000,   // tile_dim0=1 (bits 127:112)
                     (int)0x00000800,   // tile_dim1=2048 (bits 143:128)
                     (int)0x00000002,   // tensor_dim0_stride=2 (bits 191:160)
                     0, 0 };
    const v4i gz = { 0, 0, 0, 0 };
#if defined(ICD_TDM_6ARG)
    const v8i gz8 = { 0, 0, 0, 0, 0, 0, 0, 0 };
    __builtin_amdgcn_tensor_load_to_lds(g0, g1, gz, gz, gz8, 0);
#else
    __builtin_amdgcn_tensor_load_to_lds(g0, g1, gz, gz, 0);
#endif
    __builtin_amdgcn_s_wait_tensorcnt(0);
  }
  __syncthreads();

  // contiguous, conflict-benign LDS reads: 2x ds_load_b128 per thread
  const float4* ap = reinterpret_cast<const float4*>(&abuf[l0]);
  const float4 a0 = ap[0];
  const float4 a1 = ap[1];
  alpha[0] = a0.x; alpha[1] = a0.y; alpha[2] = a0.z; alpha[3] = a0.w;
  alpha[4] = a1.x; alpha[5] = a1.y; alpha[6] = a1.z; alpha[7] = a1.w;
#else
  // ---- fallback: direct strided global loads of channel 0 ----
#pragma unroll
  for (int i = 0; i < kEPT; ++i) alpha[i] = row[(size_t)(l0 + i) * 2];
#endif

  unsigned char mb[kEPT];
#pragma unroll
  for (int i = 0; i < kEPT; ++i) mb[i] = mrow[l0 + i];

  // ---- thread-local inclusive scan over 8 elements ----
  S loc[kEPT];
  loc[0] = elem(alpha[0], l0 + 0);
#pragma unroll
  for (int i = 1; i < kEPT; ++i) loc[i] = comb(loc[i - 1], elem(alpha[i], l0 + i));

  // ---- wave32 inclusive scan of thread totals (ds_bpermute shuffles) ----
  S ws = loc[kEPT - 1];
#pragma unroll
  for (int off = 1; off < 32; off <<= 1) {
    const S up = shfl_up_state(ws, off);
    if (lane >= off) ws = comb(up, ws);
  }
  // wave-exclusive prefix for this thread
  S wprev = shfl_up_state(ws, 1);
  if (lane == 0) wprev = ident();

  // ---- cross-wave scan via LDS (8 wave totals) ----
  __shared__ S wtot[kTPB / 32];
  if (lane == 31) wtot[wid] = ws;
  __syncthreads();
  S bpre = ident();
#pragma unroll
  for (int i = 0; i < kTPB / 32; ++i) {
    if (i < wid) bpre = comb(bpre, wtot[i]);
  }

  // total exclusive prefix for this thread (earlier waves, then earlier lanes)
  const S tex = comb(bpre, wprev);

  // ---- emit coordinates ----
  float ob[kEPT * 3];
#pragma unroll
  for (int i = 0; i < kEPT; ++i) {
    const int l = l0 + i;
    float x, y;
    if (l == 0)      { x = 0.f;  y = 0.f;  }
    else if (l == 1) { x = kBL;  y = 0.f;  }
    else if (l == 2) { x = kBL;  y = kBL;  }
    else {
      const S g = comb(tex, loc[i]);   // inclusive scan value at l
      x = kBL + kBL * g.y;
      y = kBL - kBL * g.x;
    }
    const float m = mb[i] ? 1.f : 0.f;
    ob[3 * i + 0] = x * m;
    ob[3 * i + 1] = y * m;
    ob[3 * i + 2] = 0.f;
  }

  // 96 contiguous bytes per thread, 16B-aligned -> 6x b128 stores
  float4* op = reinterpret_cast<float4*>(out + ((size_t)b * kL + l0) * 3);
  const float4* os = reinterpret_cast<const float4*>(ob);
#pragma unroll
  for (int i = 0; i < (kEPT * 3) / 4; ++i) op[i] = os[i];
}

extern "C" void kernel_launch(void* const* d_in, const int* in_sizes, int n_in,
                              void* d_out, int out_size, void* d_ws, size_t ws_size,
                              hipStream_t stream) {
  (void)n_in; (void)out_size; (void)d_ws; (void)ws_size;
  const float* torsions = reinterpret_cast<const float*>(d_in[0]);        // [B,L,2] f32
  const unsigned char* mask = reinterpret_cast<const unsigned char*>(d_in[1]);  // [B,L] bool
  float* out = reinterpret_cast<float*>(d_out);                           // [B,L,3] f32

  const int B = in_sizes[1] / kL;   // mask flat count = B*L
  if (B <= 0) return;
  icd_scan_kernel<<<B, kTPB, 0, stream>>>(torsions, mask, out);
}